// ConditionedPNA_87076166959718
// MI455X (gfx1250) — compile-verified
//
#include <hip/hip_runtime.h>
#include <hip/hip_bf16.h>
#include <math.h>

// Problem constants (from reference)
#define BB 4
#define NN 10000
#define EE 80000
#define DD 128
#define KQ 8
#define RR 237
#define LAYERS 3
#define BNODES (BB * NN)          // 40000
#define KSEL 1000                 // max(1, int(0.1*N))
#define EPSV 1e-5f
#define NT 4                      // N-tiles per wave (64 output cols)

static_assert(BNODES % 16 == 0, "M must be multiple of 16 for WMMA tiles");
static_assert(DD % (16 * NT) == 0, "col tiling");

typedef __attribute__((ext_vector_type(2))) float v2f;
typedef __attribute__((ext_vector_type(8))) float v8f;

// ---------------- order-preserving float <-> uint key (for atomic max/min) ---
__device__ __forceinline__ unsigned fkey(float f) {
    unsigned u = __float_as_uint(f);
    return (u & 0x80000000u) ? ~u : (u | 0x80000000u);
}
__device__ __forceinline__ float fdec(unsigned k) {
    unsigned u = (k & 0x80000000u) ? (k & 0x7fffffffu) : ~k;
    return __uint_as_float(u);
}

// ---------------- generic WMMA GEMM: Out[M x 128] = A[M x K] @ W^T (+bias) ---
// W is row-major [128 x K] with row stride ldw (supports packed submatrices).
// A is produced by a mode-specific loader so concats/products are never
// materialized in HBM.
struct APar {
    const float* A0;
    const float* A1;
    const float* bmean;
    const float* bmax;
    const float* bmin;
    const float* bstd;
    const float* logd;
    const float* invld;
};

// MODE 0: plain A0[n, i]                       (K = 128)
// MODE 1: concat(A0[n, :], rel[A_batch(n), :]) (K = 256)
// MODE 2: A0[n,i] * A1[n,i]                    (K = 128)
// MODE 3: concat(hidden, PNA agg * deg scales) (K = 1664)
template <int MODE>
__device__ __forceinline__ float loadA(const APar& p, int n, int i) {
    if (MODE == 0) return p.A0[(size_t)n * DD + i];
    if (MODE == 1) {
        return (i < DD) ? p.A0[(size_t)n * DD + i]
                        : p.A1[(size_t)(n / NN) * DD + (i - DD)];
    }
    if (MODE == 2) return p.A0[(size_t)n * DD + i] * p.A1[(size_t)n * DD + i];
    // MODE 3 — post aggregate: [hidden(128) | base(512) | base*logd | base/logd]
    if (i < DD) return p.A0[(size_t)n * DD + i];
    int j = i - DD;
    int t = j >> 9;          // which scale block (0..2)
    int jj = j & 511;
    int c = jj >> 7;         // 0=mean 1=max 2=min 3=std
    int d = jj & 127;
    const float* base = (c == 0) ? p.bmean : (c == 1) ? p.bmax
                       : (c == 2) ? p.bmin : p.bstd;
    float v = base[(size_t)n * DD + d];
    float s = (t == 0) ? 1.0f : (t == 1) ? p.logd[n] : p.invld[n];
    return v * s;
}

// Each wave: one 16-row M-tile x NT 16-col N-tiles. A fragment is loaded once
// per K-step and feeds NT back-to-back v_wmma issues (better XDL utilization,
// NT x less A traffic through L2).
template <int MODE, bool RELU>
__global__ __launch_bounds__(256) void wmma_gemm_kernel(
    APar ap, const float* __restrict__ W, const float* __restrict__ bias,
    float* __restrict__ Out, int Kdim, int ldw) {
    const int Mtiles = BNODES / 16;
    int wave = threadIdx.x >> 5;
    int lane = threadIdx.x & 31;
    int mtile = blockIdx.y * 8 + wave;   // wave-uniform guard (EXEC stays full)
    if (mtile >= Mtiles) return;
    int l16 = lane & 15;
    int kh = lane >> 4;                  // which K half-pair this lane holds
    int row = mtile * 16 + l16;          // A row for this lane
    int col0 = blockIdx.x * (16 * NT) + l16;

    const float* wr[NT];
#pragma unroll
    for (int nt = 0; nt < NT; ++nt)
        wr[nt] = W + (size_t)(col0 + nt * 16) * ldw;

    v8f acc[NT];
#pragma unroll
    for (int nt = 0; nt < NT; ++nt) acc[nt] = (v8f){0.f, 0.f, 0.f, 0.f,
                                                    0.f, 0.f, 0.f, 0.f};

    for (int k0 = 0; k0 < Kdim; k0 += 4) {
        int ka = k0 + kh * 2;
        // A 16x4 fragment: lanes 0-15 hold K={k0,k0+1}, lanes 16-31 K={k0+2,k0+3}
        float a0 = loadA<MODE>(ap, row, ka);
        float a1 = loadA<MODE>(ap, row, ka + 1);
        v2f av; av.x = a0; av.y = a1;
#pragma unroll
        for (int nt = 0; nt < NT; ++nt) {
            // B 4x16 fragment (B[k][col] = W[col][k]), mirrored lane layout
            v2f bv; bv.x = wr[nt][ka]; bv.y = wr[nt][ka + 1];
            acc[nt] = __builtin_amdgcn_wmma_f32_16x16x4_f32(
                false, av, false, bv, (short)0, acc[nt], false, false);
        }
    }
#pragma unroll
    for (int nt = 0; nt < NT; ++nt) {
        int col = col0 + nt * 16;
        float bvv = bias ? bias[col] : 0.f;
#pragma unroll
        for (int r = 0; r < 8; ++r) {
            int orow = mtile * 16 + kh * 8 + r;  // lanes 16-31 -> M rows 8..15
            float v = acc[nt][r] + bvv;
            if (RELU) v = fmaxf(v, 0.f);
            Out[(size_t)orow * DD + col] = v;
        }
    }
}

// ---------------- setup: per-batch head/tail/relation resolution -------------
__global__ void setup_kernel(const int* __restrict__ h_index,
                             const int* __restrict__ r_index,
                             const int* __restrict__ t_index,
                             int* hg, int* tg, int* tailo, int* relrow) {
    if (blockIdx.x == 0 && threadIdx.x == 0) {
        for (int b = 0; b < BB; ++b) {
            bool tn = true;
            for (int k = 1; k < KQ; ++k)
                if (h_index[b * KQ + k] != h_index[b * KQ]) tn = false;
            int h0 = tn ? h_index[b * KQ] : t_index[b * KQ];
            int t0 = tn ? t_index[b * KQ] : h_index[b * KQ];
            int r0 = tn ? r_index[b * KQ] : r_index[b * KQ] + RR;
            hg[b] = b * NN + h0;
            tg[b] = b * NN + t0;
            tailo[b] = t0;
            relrow[b] = r0;
        }
    }
}

// ---------------- initial scores: bg (hidden=0) and head-node score ----------
__global__ __launch_bounds__(128) void score0_kernel(
    const float* __restrict__ hidden_states, const float* __restrict__ rel_emb,
    const int* __restrict__ relrow, const float* __restrict__ linW,
    const float* __restrict__ linb, const float* __restrict__ m1W,
    const float* __restrict__ m1b, const float* __restrict__ m2W,
    const float* __restrict__ m2b, float* __restrict__ relbuf,
    float* __restrict__ bg, float* __restrict__ hscore) {
    int b = blockIdx.x;
    int o = threadIdx.x;
    __shared__ float sh[DD], sr[DD], sx[DD], red[DD];
    sh[o] = hidden_states[b * DD + o];
    float rv = rel_emb[(size_t)relrow[b] * DD + o];
    sr[o] = rv;
    relbuf[b * DD + o] = rv;
    __syncthreads();
    float acc = linb[o];
    for (int i = 0; i < DD; ++i) acc += sh[i] * linW[(size_t)o * 256 + i];
    for (int i = 0; i < DD; ++i) acc += sr[i] * linW[(size_t)o * 256 + 128 + i];
    sx[o] = acc * sh[o];
    __syncthreads();
    float acc2 = m1b[o];
    for (int i = 0; i < DD; ++i) acc2 += sx[i] * m1W[(size_t)o * DD + i];
    float y = fmaxf(acc2, 0.f);
    red[o] = y * m2W[o];
    __syncthreads();
    for (int st = 64; st > 0; st >>= 1) {
        if (o < st) red[o] += red[o + st];
        __syncthreads();
    }
    if (o == 0) hscore[b] = red[0] + m2b[0];
    __syncthreads();
    // bg = score(hidden=0): x = 0 -> y = relu(m1b)
    red[o] = fmaxf(m1b[o], 0.f) * m2W[o];
    __syncthreads();
    for (int st = 64; st > 0; st >>= 1) {
        if (o < st) red[o] += red[o + st];
        __syncthreads();
    }
    if (o == 0 && b == 0) bg[0] = red[0] + m2b[0];
}

// ---------------- x_rep / initial score init ---------------------------------
__global__ __launch_bounds__(256) void init_x_kernel(
    const float* __restrict__ graph_x, const float* __restrict__ score_text,
    const float* __restrict__ hidden_states, const int* __restrict__ hg,
    const int* __restrict__ tg, const int* __restrict__ tailo,
    const float* __restrict__ bg, const float* __restrict__ hscore,
    float* __restrict__ hidden, float* __restrict__ score) {
    int idx = blockIdx.x * blockDim.x + threadIdx.x;
    if (idx >= BNODES * DD) return;
    int n = idx >> 7;
    int d = idx & 127;
    int b = n / NN;
    int ln = n - b * NN;
    float v = graph_x[(size_t)ln * DD + d];
    if (n == tg[b]) v = score_text[(size_t)tailo[b] * DD + d];
    if (n == hg[b]) v = hidden_states[b * DD + d];   // head overrides tail
    hidden[idx] = v;
    if (d == 0) {
        float s = bg[0];
        if (n == hg[b]) s = hscore[b];
        score[n] = s;
    }
}

// ---------------- per-batch top-k threshold (radix bisection in LDS) ---------
__global__ __launch_bounds__(1024) void thresh_kernel(
    const float* __restrict__ score, float* __restrict__ thresh) {
    __shared__ unsigned skeys[NN];   // 40KB of the 320KB/WGP LDS
    __shared__ int cnt;
    int b = blockIdx.x;
    int tid = threadIdx.x;
    for (int i = tid; i < NN; i += 1024) skeys[i] = fkey(score[b * NN + i]);
    __syncthreads();
    unsigned cur = 0u;
    for (int bit = 31; bit >= 0; --bit) {
        unsigned trial = cur | (1u << bit);
        if (tid == 0) cnt = 0;
        __syncthreads();
        int local = 0;
        for (int i = tid; i < NN; i += 1024) local += (skeys[i] >= trial) ? 1 : 0;
        atomicAdd(&cnt, local);
        __syncthreads();
        if (cnt >= KSEL) cur = trial;    // keep largest key with count >= k
        __syncthreads();
    }
    if (tid == 0) thresh[b] = fdec(cur);
}

// ---------------- accumulator reset ------------------------------------------
__global__ __launch_bounds__(256) void init_acc_kernel(
    float* asum, float* asq, unsigned* amax, unsigned* amin, float* deg,
    float* odeg) {
    int idx = blockIdx.x * blockDim.x + threadIdx.x;
    if (idx >= BNODES * DD) return;
    asum[idx] = 0.f;
    asq[idx] = 0.f;
    amax[idx] = 0u;            // key of -max float
    amin[idx] = 0xffffffffu;   // key of +max float
    if ((idx & 127) == 0) {
        int n = idx >> 7;
        deg[n] = 0.f;
        odeg[n] = 0.f;
    }
}

// ---------------- edge aggregation: one warp per selected edge ---------------
__global__ __launch_bounds__(256) void edge_agg_kernel(
    const int* __restrict__ edge, const float* __restrict__ score,
    const float* __restrict__ thresh, const float* __restrict__ pi,
    const float* __restrict__ pj, float* __restrict__ asum,
    float* __restrict__ asq, unsigned* __restrict__ amax,
    unsigned* __restrict__ amin, float* __restrict__ deg,
    float* __restrict__ odeg) {
    int gw = (blockIdx.x * blockDim.x + threadIdx.x) >> 5;
    int lane = threadIdx.x & 31;
    if (gw >= BB * EE) return;                       // warp-uniform
    int b = gw / EE;
    int e = gw - b * EE;
    int s = edge[e];
    int t = edge[EE + e];
    if (!(score[b * NN + s] >= thresh[b])) return;   // emask, warp-uniform
    int sg = b * NN + s;
    int tg = b * NN + t;
    if (lane == 0) {
        atomicAdd(&deg[tg], 1.f);
        atomicAdd(&odeg[sg], 1.f);
    }
    const float* pit = pi + (size_t)tg * DD;   // proj_i[tgt] (preb folded in)
    const float* pjs = pj + (size_t)sg * DD;   // proj_j[src]
#pragma unroll
    for (int j = 0; j < 4; ++j) {
        int d = lane + 32 * j;
        float m = pit[d] + pjs[d];
        size_t o = (size_t)tg * DD + d;
        atomicAdd(&asum[o], m);
        atomicAdd(&asq[o], m * m);
        unsigned k = fkey(m);
        atomicMax(&amax[o], k);
        atomicMin(&amin[o], k);
    }
}

// ---------------- finalize PNA statistics in place ---------------------------
__global__ __launch_bounds__(256) void finalize_kernel(
    float* __restrict__ asum, float* __restrict__ asq,
    unsigned* __restrict__ amax, unsigned* __restrict__ amin,
    const float* __restrict__ deg, float* __restrict__ logd,
    float* __restrict__ invld) {
    int idx = blockIdx.x * blockDim.x + threadIdx.x;
    if (idx >= BNODES * DD) return;
    int n = idx >> 7;
    int d = idx & 127;
    float dv = deg[n];
    float dc = fmaxf(dv, 1.f);
    float mean = asum[idx] / dc;
    float msq = asq[idx] / dc;
    float var = msq - mean * mean;
    if (var < 0.f) var = 0.f;
    float stdv = sqrtf(var + EPSV);
    float mx = (dv > 0.f) ? fdec(amax[idx]) : 0.f;
    float mn = (dv > 0.f) ? fdec(amin[idx]) : 0.f;
    asum[idx] = mean;
    asq[idx] = stdv;
    ((float*)amax)[idx] = mx;
    ((float*)amin)[idx] = mn;
    if (d == 0) {
        float ld = logf(dc + 1.f);
        logd[n] = ld;
        invld[n] = 1.f / ld;   // AVG_DEG_LOG == 1.0
    }
}

// ---------------- residual update gated by out-degree ------------------------
__global__ __launch_bounds__(256) void update_hidden_kernel(
    float* __restrict__ hidden, const float* __restrict__ nh,
    const float* __restrict__ odeg) {
    int idx = blockIdx.x * blockDim.x + threadIdx.x;
    if (idx >= BNODES * DD) return;
    int n = idx >> 7;
    if (odeg[n] > 0.f) hidden[idx] += nh[idx];
}

// ---------------- final score: warp-per-node dot with mlp2 -------------------
__global__ __launch_bounds__(256) void score_dot_kernel(
    const float* __restrict__ y, const float* __restrict__ m2W,
    const float* __restrict__ m2b, float* __restrict__ score) {
    int gw = (blockIdx.x * blockDim.x + threadIdx.x) >> 5;
    int lane = threadIdx.x & 31;
    if (gw >= BNODES) return;
    const float* yr = y + (size_t)gw * DD;
    float s = 0.f;
#pragma unroll
    for (int j = 0; j < 4; ++j) s += yr[lane + 32 * j] * m2W[lane + 32 * j];
    for (int off = 16; off > 0; off >>= 1) s += __shfl_down(s, off, 32);
    if (lane == 0) score[gw] = s + m2b[0];
}

// ---------------- gather output ----------------------------------------------
__global__ void out_kernel(const float* __restrict__ score,
                           const int* __restrict__ tg, float* __restrict__ out) {
    int b = threadIdx.x;
    if (b < BB) out[b] = score[tg[b]];
}

extern "C" void kernel_launch(void* const* d_in, const int* in_sizes, int n_in,
                              void* d_out, int out_size, void* d_ws,
                              size_t ws_size, hipStream_t stream) {
    (void)in_sizes; (void)n_in; (void)out_size; (void)ws_size;

    const int* h_index = (const int*)d_in[0];
    const int* r_index = (const int*)d_in[1];
    const int* t_index = (const int*)d_in[2];
    const float* hidden_states = (const float*)d_in[3];
    const float* graph_x = (const float*)d_in[5];
    const int* edge_index = (const int*)d_in[6];
    const float* score_text_embs = (const float*)d_in[7];
    // params flattened in jax pytree (sorted-key) order:
    // layers[l]: linW, linb, postW, postb, preW, preb ; then top-level
    const float *lw[LAYERS], *lb[LAYERS], *postW[LAYERS], *postb[LAYERS],
        *preW[LAYERS], *preb[LAYERS];
    int p = 9;
    for (int l = 0; l < LAYERS; ++l) {
        lw[l] = (const float*)d_in[p + 0];
        lb[l] = (const float*)d_in[p + 1];
        postW[l] = (const float*)d_in[p + 2];
        postb[l] = (const float*)d_in[p + 3];
        preW[l] = (const float*)d_in[p + 4];
        preb[l] = (const float*)d_in[p + 5];
        p += 6;
    }
    const float* linW = (const float*)d_in[p + 0];
    const float* linb = (const float*)d_in[p + 1];
    const float* m1W = (const float*)d_in[p + 2];
    const float* m1b = (const float*)d_in[p + 3];
    const float* m2W = (const float*)d_in[p + 4];
    const float* m2b = (const float*)d_in[p + 5];
    const float* rel_emb = (const float*)d_in[p + 6];

    float* out = (float*)d_out;

    // workspace carve-up
    float* w = (float*)d_ws;
    const size_t ND = (size_t)BNODES * DD;
    float* hidden = w; w += ND;
    float* tmp1 = w; w += ND;            // proj_i, later post output
    float* tmp2 = w; w += ND;            // proj_j, later lin output
    float* bmean = w; w += ND;           // sum -> mean, later heur
    float* bstd = w; w += ND;            // sumsq -> std, later y
    unsigned* bmax = (unsigned*)w; w += ND;
    unsigned* bmin = (unsigned*)w; w += ND;
    float* score = w; w += BNODES;
    float* deg = w; w += BNODES;
    float* odeg = w; w += BNODES;
    float* logd = w; w += BNODES;
    float* invld = w; w += BNODES;
    float* thresh = w; w += 16;
    float* relbuf = w; w += BB * DD;
    float* bg = w; w += 16;
    float* hscore = w; w += 16;
    int* hg = (int*)w;
    int* tg = hg + 4;
    int* tailo = hg + 8;
    int* relrow = hg + 12;

    const dim3 gemmGrid(DD / (16 * NT), (BNODES / 16 + 7) / 8);  // (2, 313)
    const int EW = ND;                                    // elementwise size
    const int ewBlocks = (EW + 255) / 256;                // 20000
    const int edgeBlocks = (BB * EE * 32 + 255) / 256;    // 40000
    const int dotBlocks = (BNODES * 32 + 255) / 256;      // 5000

    setup_kernel<<<1, 32, 0, stream>>>(h_index, r_index, t_index, hg, tg, tailo,
                                       relrow);
    score0_kernel<<<BB, 128, 0, stream>>>(hidden_states, rel_emb, relrow, linW,
                                          linb, m1W, m1b, m2W, m2b, relbuf, bg,
                                          hscore);
    init_x_kernel<<<ewBlocks, 256, 0, stream>>>(graph_x, score_text_embs,
                                                hidden_states, hg, tg, tailo,
                                                bg, hscore, hidden, score);

    for (int l = 0; l < LAYERS; ++l) {
        thresh_kernel<<<BB, 1024, 0, stream>>>(score, thresh);
        init_acc_kernel<<<ewBlocks, 256, 0, stream>>>(bmean, bstd, bmax, bmin,
                                                      deg, odeg);
        APar apH = {hidden, nullptr, nullptr, nullptr, nullptr, nullptr,
                    nullptr, nullptr};
        // proj_i = hidden @ preW[:, :D].T + preb (bias folded here)
        wmma_gemm_kernel<0, false><<<gemmGrid, 256, 0, stream>>>(
            apH, preW[l], preb[l], tmp1, DD, 2 * DD);
        // proj_j = hidden @ preW[:, D:].T
        wmma_gemm_kernel<0, false><<<gemmGrid, 256, 0, stream>>>(
            apH, preW[l] + DD, nullptr, tmp2, DD, 2 * DD);
        edge_agg_kernel<<<edgeBlocks, 256, 0, stream>>>(
            edge_index, score, thresh, tmp1, tmp2, bmean, bstd, bmax, bmin, deg,
            odeg);
        finalize_kernel<<<ewBlocks, 256, 0, stream>>>(bmean, bstd, bmax, bmin,
                                                      deg, logd, invld);
        // post: [hidden | agg13] @ postW.T + postb   (K = 1664, never materialized)
        APar apP = {hidden, nullptr, bmean, (const float*)bmax,
                    (const float*)bmin, bstd, logd, invld};
        wmma_gemm_kernel<3, false><<<gemmGrid, 256, 0, stream>>>(
            apP, postW[l], postb[l], tmp1, 13 * DD, 13 * DD);
        // layer lin
        APar apT = {tmp1, nullptr, nullptr, nullptr, nullptr, nullptr, nullptr,
                    nullptr};
        wmma_gemm_kernel<0, false><<<gemmGrid, 256, 0, stream>>>(
            apT, lw[l], lb[l], tmp2, DD, DD);
        update_hidden_kernel<<<ewBlocks, 256, 0, stream>>>(hidden, tmp2, odeg);
        // score = _score(hidden, rel_per_node)
        APar apC = {hidden, relbuf, nullptr, nullptr, nullptr, nullptr, nullptr,
                    nullptr};
        wmma_gemm_kernel<1, false><<<gemmGrid, 256, 0, stream>>>(
            apC, linW, linb, bmean /*heur*/, 2 * DD, 2 * DD);
        APar apM = {bmean /*heur*/, hidden, nullptr, nullptr, nullptr, nullptr,
                    nullptr, nullptr};
        wmma_gemm_kernel<2, true><<<gemmGrid, 256, 0, stream>>>(
            apM, m1W, m1b, bstd /*y*/, DD, DD);
        score_dot_kernel<<<dotBlocks, 256, 0, stream>>>(bstd, m2W, m2b, score);
    }

    out_kernel<<<1, 32, 0, stream>>>(score, tg, out);
}